// DetBenchEvalMultiScale_75788992906164
// MI455X (gfx1250) — compile-verified
//
#include <hip/hip_runtime.h>
#include <math.h>

#define NANCH  196416
#define NCAND  5000
#define NDET   100
#define BATCHN 32
#define TIEBUF 8192

#if defined(__HIP_DEVICE_COMPILE__) && __has_builtin(__builtin_amdgcn_tensor_load_to_lds) && __has_builtin(__builtin_amdgcn_s_wait_tensorcnt)
#define USE_TDM 1
#else
#define USE_TDM 0
#endif

#if USE_TDM
typedef unsigned int u32x4 __attribute__((ext_vector_type(4)));
typedef int          i32x4 __attribute__((ext_vector_type(4)));
typedef int          i32x8 __attribute__((ext_vector_type(8)));
#endif

// Order-preserving float32 -> uint32 key (ascending key == ascending float).
__device__ __forceinline__ unsigned f2key(float f) {
  unsigned u = __float_as_uint(f);
  return u ^ ((u & 0x80000000u) ? 0xFFFFFFFFu : 0x80000000u);
}

// ---------------------------------------------------------------------------
// Kernel A: per-image exact top-5000 selection.
// 4-pass MSB radix select on 32-bit keys finds the exact 5000th-largest key.
// Ties at that key are resolved by smallest global anchor index g (jax rule).
// Finally compacts the exactly-5000 winners (anchor index + logit) to ws.
// One 1024-thread (32-wave) workgroup per image.
// ---------------------------------------------------------------------------
__global__ __launch_bounds__(1024) void topk_select(
    const float* __restrict__ c0, const float* __restrict__ c1,
    const float* __restrict__ c2, const float* __restrict__ c3,
    const float* __restrict__ c4,
    int* __restrict__ cand_g, float* __restrict__ cand_logit)
{
  const int b  = blockIdx.x;
  const int t  = threadIdx.x;
  const int NT = blockDim.x;

  __shared__ unsigned hist[256];
  __shared__ unsigned tiebuf[TIEBUF];
  __shared__ unsigned sh_prefix, sh_remaining, sh_tieT, sh_gstar, sh_slot;

  const int lsz[5]  = {9 * 128 * 128, 9 * 64 * 64, 9 * 32 * 32, 9 * 16 * 16, 9 * 8 * 8};
  const int ssz[5]  = {128 * 128, 64 * 64, 32 * 32, 16 * 16, 8 * 8};
  const int goff[5] = {0, 147456, 184320, 193536, 195840};
  const float* lp[5];
  lp[0] = c0 + (size_t)b * lsz[0];
  lp[1] = c1 + (size_t)b * lsz[1];
  lp[2] = c2 + (size_t)b * lsz[2];
  lp[3] = c3 + (size_t)b * lsz[3];
  lp[4] = c4 + (size_t)b * lsz[4];

  if (t == 0) { sh_prefix = 0u; sh_remaining = NCAND; sh_gstar = 0u; }
  __syncthreads();

  // ---- 4 radix passes, MSB first ----
  for (int p = 3; p >= 0; --p) {
    for (int i = t; i < 256; i += NT) hist[i] = 0u;
    __syncthreads();
    const unsigned prefix   = sh_prefix;
    const unsigned maskHigh = (p == 3) ? 0u : (0xFFFFFFFFu << ((p + 1) * 8));
    const int      shift    = p * 8;
    for (int l = 0; l < 5; ++l) {
      const float* __restrict__ pp = lp[l];
      const int n = lsz[l];
      for (int r = t; r < n; r += NT) {
        if (p == 3 && (r + 8192) < n) __builtin_prefetch(pp + r + 8192, 0, 0); // global_prefetch_b8
        unsigned key = f2key(pp[r]);
        if ((key & maskHigh) == (prefix & maskHigh))
          atomicAdd(&hist[(key >> shift) & 255u], 1u);
      }
    }
    __syncthreads();
    if (t == 0) {
      unsigned remaining = sh_remaining;
      unsigned cum = 0u;
      int bbin = 0;
      for (int i2 = 255; i2 >= 0; --i2) {
        unsigned h = hist[i2];
        if (cum + h >= remaining) { bbin = i2; break; }
        cum += h;
      }
      sh_prefix    = prefix | ((unsigned)bbin << shift);
      sh_remaining = remaining - cum;   // rank of target within the tie-bin
    }
    __syncthreads();
  }

  const unsigned kth = sh_prefix;       // exact 5000th-largest key
  const unsigned m   = sh_remaining;    // how many key==kth elements to keep (>=1)

  // ---- tie pass: gather anchor indices of all elements equal to kth key ----
  if (t == 0) sh_tieT = 0u;
  __syncthreads();
  for (int l = 0; l < 5; ++l) {
    const float* __restrict__ pp = lp[l];
    const int n = lsz[l], SS = ssz[l], go = goff[l];
    for (int r = t; r < n; r += NT) {
      if (f2key(pp[r]) == kth) {
        unsigned pos = atomicAdd(&sh_tieT, 1u);
        int k = r / SS; int cell = r - k * SS;      // raw NCHW -> (cell, k)
        if (pos < TIEBUF) tiebuf[pos] = (unsigned)(go + cell * 9 + k);
      }
    }
  }
  __syncthreads();
  unsigned T = sh_tieT; if (T > TIEBUF) T = TIEBUF;
  // m-th smallest g among ties (g values are distinct): count-based select
  for (unsigned i = t; i < T; i += NT) {
    unsigned gi = tiebuf[i];
    unsigned c = 0u;
    for (unsigned j = 0; j < T; ++j) c += (tiebuf[j] < gi) ? 1u : 0u;
    if (c == m - 1u) sh_gstar = gi;
  }
  __syncthreads();
  const unsigned gstar = sh_gstar;

  // ---- compaction: key > kth, or key == kth with g <= gstar (exactly 5000) ----
  if (t == 0) sh_slot = 0u;
  __syncthreads();
  int*   cg = cand_g     + (size_t)b * NCAND;
  float* cl = cand_logit + (size_t)b * NCAND;
  for (int l = 0; l < 5; ++l) {
    const float* __restrict__ pp = lp[l];
    const int n = lsz[l], SS = ssz[l], go = goff[l];
    for (int r = t; r < n; r += NT) {
      float f = pp[r];
      unsigned key = f2key(f);
      if (key < kth) continue;
      int k = r / SS; int cell = r - k * SS;
      unsigned g = (unsigned)(go + cell * 9 + k);
      if (key > kth || g <= gstar) {
        unsigned slot = atomicAdd(&sh_slot, 1u);
        if (slot < NCAND) { cg[slot] = (int)g; cl[slot] = f; }
      }
    }
  }
}

// ---------------------------------------------------------------------------
// Kernel B: decode the 32x5000 candidates. Anchors rebuilt in fp64 (matching
// NumPy) then rounded to fp32; box regressions gathered from NCHW inputs.
// Writes 6 of the 8 SoA rows the NMS kernel DMA-loads as one 2D tensor tile.
// ---------------------------------------------------------------------------
__global__ void decode_cand(
    const float* __restrict__ b0, const float* __restrict__ b1,
    const float* __restrict__ b2, const float* __restrict__ b3,
    const float* __restrict__ b4,
    const int* __restrict__ cand_g, const float* __restrict__ cand_logit,
    float* __restrict__ ox1, float* __restrict__ oy1,
    float* __restrict__ ox2, float* __restrict__ oy2,
    float* __restrict__ oarea, float* __restrict__ oscore)
{
  int idx = blockIdx.x * blockDim.x + threadIdx.x;
  if (idx >= BATCHN * NCAND) return;
  int b = idx / NCAND;
  int g = cand_g[idx];

  int lvl, go, S, stride;
  if      (g < 147456) { lvl = 0; go = 0;      S = 128; stride = 8;   }
  else if (g < 184320) { lvl = 1; go = 147456; S = 64;  stride = 16;  }
  else if (g < 193536) { lvl = 2; go = 184320; S = 32;  stride = 32;  }
  else if (g < 195840) { lvl = 3; go = 193536; S = 16;  stride = 64;  }
  else                 { lvl = 4; go = 195840; S = 8;   stride = 128; }

  int local = g - go;
  int cell  = local / 9;
  int k     = local - cell * 9;
  int y     = cell / S;
  int x     = cell - y * S;
  int o     = k / 3;
  int ai    = k - o * 3;

  const double OCT[3]  = {1.0, 1.2599210498948731647672106, 1.5874010519681993613971827};
  const double ASPX[3] = {1.0, 1.4, 0.7};
  const double ASPY[3] = {1.0, 0.7, 1.4};
  double base = 4.0 * (double)stride * OCT[o];
  double ax2  = base * ASPX[ai] * 0.5;
  double ay2  = base * ASPY[ai] * 0.5;
  double xv   = 0.5 * (double)stride + (double)x * (double)stride;
  double yv   = 0.5 * (double)stride + (double)y * (double)stride;
  float a0 = (float)(yv - ay2);   // y1a
  float a1 = (float)(xv - ax2);   // x1a
  float a2 = (float)(yv + ay2);   // y2a
  float a3 = (float)(xv + ax2);   // x2a
  float yca = (a0 + a2) * 0.5f;
  float xca = (a1 + a3) * 0.5f;
  float ha  = a2 - a0;
  float wa  = a3 - a1;

  const float* bp;
  switch (lvl) { case 0: bp = b0; break; case 1: bp = b1; break;
                 case 2: bp = b2; break; case 3: bp = b3; break; default: bp = b4; }
  int SS = S * S;
  const float* bb = bp + (size_t)b * 36 * SS + (size_t)(k * 4) * SS + cell;
  float ty = bb[0];
  float tx = bb[SS];
  float th = bb[2 * SS];
  float tw = bb[3 * SS];

  float w  = expf(tw) * wa;
  float h  = expf(th) * ha;
  float yc = ty * ha + yca;
  float xc = tx * wa + xca;
  float X1 = xc - w * 0.5f;
  float Y1 = yc - h * 0.5f;
  float X2 = xc + w * 0.5f;
  float Y2 = yc + h * 0.5f;

  ox1[idx] = X1; oy1[idx] = Y1; ox2[idx] = X2; oy2[idx] = Y2;
  oarea[idx] = (X2 - X1) * (Y2 - Y1);
  float lg = cand_logit[idx];
  oscore[idx] = 1.0f / (1.0f + expf(-lg));
}

// ---------------------------------------------------------------------------
// Kernel C: greedy NMS, one 1024-thread WG per image. The full candidate set
// (160 KB: 8 rows x 5000 f32) is brought into LDS by ONE Tensor-Data-Mover
// descriptor: 2D tile 5000x8, global row stride 160000 elements (the SoA
// array pitch in the workspace), packed contiguously into LDS. Wave 0 issues
// tensor_load_to_lds, waits s_wait_tensorcnt 0, then the workgroup barrier
// publishes LDS to all 32 waves. CDNA5's 320KB LDS holds the whole problem.
// Argmax runs on raw logits with anchor-index tie-break, which exactly
// reproduces jax.lax.top_k ordering + jnp.argmax first-occurrence semantics
// (sigmoid is monotone but not fp32-injective), and makes the output invariant
// to the nondeterministic compaction order.
// ---------------------------------------------------------------------------
__global__ __launch_bounds__(1024) void nms_kernel(
    const float* __restrict__ wsbase,   // = cand_g base; 8 rows of 160000 f32
    const float* __restrict__ scales, float* __restrict__ out)
{
  __shared__ float smem[8 * NCAND];   // rows: g, logit, x1, y1, x2, y2, area, score
  __shared__ float rval[1024];
  __shared__ int   ridx[1024];
  __shared__ int   rg[1024];

  const int b = blockIdx.x, t = threadIdx.x;
  const int NT = blockDim.x;

  int*   sg  = (int*)smem;            // row 0
  float* slg = smem + 1 * NCAND;      // row 1 (suppression target)
  float* sx1 = smem + 2 * NCAND;
  float* sy1 = smem + 3 * NCAND;
  float* sx2 = smem + 4 * NCAND;
  float* sy2 = smem + 5 * NCAND;
  float* sar = smem + 6 * NCAND;
  float* ssc = smem + 7 * NCAND;

#if USE_TDM
  if (t == 0) {
    unsigned long long ga = (unsigned long long)(uintptr_t)(wsbase + (size_t)b * NCAND);
    unsigned ldsa = (unsigned)(uintptr_t)(void*)smem;   // low 32 bits = LDS byte offset
    u32x4 g0;
    g0.x = 1u;                                          // count=1 valid user D#
    g0.y = ldsa;                                        // lds_addr
    g0.z = (unsigned)(ga & 0xFFFFFFFFu);                // global_addr[31:0]
    g0.w = (unsigned)((ga >> 32) & 0x01FFFFFFu)         // global_addr[56:32]
         | (2u << 30);                                  // type = 2 ("image")
    i32x8 g1;
    g1[0] = (int)(2u << 16);                            // data_size = 4 bytes
    g1[1] = (int)((unsigned)NCAND << 16);               // tensor_dim0 = 5000 (lo16 at [63:48])
    g1[2] = (int)(8u << 16);                            // tensor_dim1 = 8 (lo16 at [95:80])
    g1[3] = (int)((unsigned)NCAND << 16);               // tile_dim0 = 5000 ([127:112])
    g1[4] = 8;                                          // tile_dim1 = 8 ([143:128])
    g1[5] = BATCHN * NCAND;                             // tensor_dim0_stride = 160000 elems
    g1[6] = 0;
    g1[7] = 0;
    i32x4 zz = {0, 0, 0, 0};                            // groups 2/3 unused (<=2D)
    i32x8 z8 = {0, 0, 0, 0, 0, 0, 0, 0};                // extra operand (clang-23 6-arg form)
    __builtin_amdgcn_tensor_load_to_lds(g0, g1, zz, zz, z8, 0);
    __builtin_amdgcn_s_wait_tensorcnt(0);
  }
#else
  for (int i = t; i < NCAND; i += NT)
    for (int rr = 0; rr < 8; ++rr)
      smem[rr * NCAND + i] = wsbase[(size_t)rr * (BATCHN * NCAND) + (size_t)b * NCAND + i];
#endif
  const float scale = scales[b];
  const float NEG = -__builtin_inff();
  __syncthreads();

  for (int it = 0; it < NDET; ++it) {
    // per-thread partial argmax on (logit desc, anchor-index asc)
    float bv = NEG; int bi = 0; int bg = 0x7FFFFFFF;
    for (int i = t; i < NCAND; i += NT) {
      float v = slg[i]; int gv = sg[i];
      if (v > bv || (v == bv && gv < bg)) { bv = v; bi = i; bg = gv; }
    }
    rval[t] = bv; ridx[t] = bi; rg[t] = bg;
    __syncthreads();
    for (int s = 512; s > 0; s >>= 1) {
      if (t < s) {
        float ov = rval[t + s]; int og = rg[t + s];
        if (ov > rval[t] || (ov == rval[t] && og < rg[t])) {
          rval[t] = ov; ridx[t] = ridx[t + s]; rg[t] = og;
        }
      }
      __syncthreads();
    }
    const int j = ridx[0];
    const float bestlg = rval[0];

    if (t == 0) {
      float* op = out + ((size_t)b * NDET + it) * 6;
      if (bestlg > NEG) {
        float X1 = sx1[j], Y1 = sy1[j], X2 = sx2[j], Y2 = sy2[j];
        op[0] = X1 * scale;
        op[1] = Y1 * scale;
        op[2] = (X2 - X1) * scale;
        op[3] = (Y2 - Y1) * scale;
        op[4] = ssc[j];             // original (unsuppressed) sigmoid score
        op[5] = 1.0f;               // class 0 + 1
      } else {
        op[0] = 0.f; op[1] = 0.f; op[2] = 0.f; op[3] = 0.f; op[4] = 0.f; op[5] = 0.f;
      }
    }

    // suppress IoU > 0.5 vs selected box (includes j itself: IoU==1)
    const float bx1 = sx1[j], by1 = sy1[j], bx2 = sx2[j], by2 = sy2[j], ba = sar[j];
    for (int i = t; i < NCAND; i += NT) {
      float xx1 = fmaxf(sx1[i], bx1), yy1 = fmaxf(sy1[i], by1);
      float xx2 = fminf(sx2[i], bx2), yy2 = fminf(sy2[i], by2);
      float inter = fmaxf(xx2 - xx1, 0.0f) * fmaxf(yy2 - yy1, 0.0f);
      float iou = inter / (sar[i] + ba - inter);
      if (iou > 0.5f) slg[i] = NEG;
    }
    __syncthreads();
  }
}

// ---------------------------------------------------------------------------
extern "C" void kernel_launch(void* const* d_in, const int* in_sizes, int n_in,
                              void* d_out, int out_size, void* d_ws, size_t ws_size,
                              hipStream_t stream)
{
  (void)in_sizes; (void)n_in; (void)out_size; (void)ws_size;
  // setup_inputs() dict order: cls_l0, box_l0, cls_l1, box_l1, ..., image_scales
  const float* cls[5]; const float* box[5];
  for (int l = 0; l < 5; ++l) {
    cls[l] = (const float*)d_in[2 * l];
    box[l] = (const float*)d_in[2 * l + 1];
  }
  const float* scales = (const float*)d_in[10];

  const int NC = BATCHN * NCAND;
  float* w       = (float*)d_ws;
  int*   cand_g  = (int*)w;            // row 0: [32*5000]
  float* cand_lg = w + 1 * NC;         // row 1
  float* cx1     = w + 2 * NC;         // row 2
  float* cy1     = w + 3 * NC;         // row 3
  float* cx2     = w + 4 * NC;         // row 4
  float* cy2     = w + 5 * NC;         // row 5
  float* car     = w + 6 * NC;         // row 6
  float* csc     = w + 7 * NC;         // row 7 -- total 5.12 MB of ws

  hipLaunchKernelGGL(topk_select, dim3(BATCHN), dim3(1024), 0, stream,
                     cls[0], cls[1], cls[2], cls[3], cls[4], cand_g, cand_lg);
  hipLaunchKernelGGL(decode_cand, dim3((NC + 255) / 256), dim3(256), 0, stream,
                     box[0], box[1], box[2], box[3], box[4],
                     cand_g, cand_lg, cx1, cy1, cx2, cy2, car, csc);
  hipLaunchKernelGGL(nms_kernel, dim3(BATCHN), dim3(1024), 0, stream,
                     w, scales, (float*)d_out);
}